// Model_41266045780632
// MI455X (gfx1250) — compile-verified
//
#include <hip/hip_runtime.h>
#include <hip/hip_bf16.h>
#include <math.h>

typedef __attribute__((ext_vector_type(2))) float v2f;
typedef __attribute__((ext_vector_type(4))) float v4f;
typedef __attribute__((ext_vector_type(8))) float v8f;

#define NB     64
#define C_IN   3
#define H_IN   512
#define W_IN   384
#define PP     16
#define QQ     12
#define OH     8
#define OW     6
#define FLATK  (C_IN * PP * QQ)   // 576
#define NOUT   512
#define THRESH 0.05f
#define GSCALE 10.0f

// ---------------------------------------------------------------------------
// Kernel 1: 32x32 average pool + threshold.  One wave per pooled element.
// Lane l streams row (p*32 + l), columns [q*32, q*32+32) as 8 float4 loads.
// grid = N*C*P blocks, 384 threads (12 waves; wave w -> q = w).
// ---------------------------------------------------------------------------
__global__ void pool_thresh_kernel(const float* __restrict__ x,
                                   float* __restrict__ thr) {
    const int b    = blockIdx.x;                 // [0, N*C*P)
    const int n    = b / (C_IN * PP);
    const int rem  = b % (C_IN * PP);
    const int c    = rem / PP;
    const int p    = rem % PP;
    const int q    = threadIdx.x >> 5;           // wave id = window col [0,12)
    const int lane = threadIdx.x & 31;

    const float* base = x
        + (size_t)((n * C_IN + c) * H_IN + p * 32 + lane) * W_IN
        + q * 32;

    float s = 0.0f;
#pragma unroll
    for (int j = 0; j < 8; ++j) {
        v4f v = *(const v4f*)(base + j * 4);
        s += v.x + v.y + v.z + v.w;
    }
    // wave32 reduction
#pragma unroll
    for (int m = 16; m >= 1; m >>= 1)
        s += __shfl_xor(s, m, 32);

    if (lane == 0) {
        float mean = s * (1.0f / 1024.0f);
        float t = (mean > THRESH) ? mean : 0.0f;
        thr[((n * C_IN + c) * PP + p) * QQ + q] = t;
    }
}

// ---------------------------------------------------------------------------
// Kernel 2: fused gate + fold.
//   gate[oh,ow] = sigmoid( (GSCALE/27) * sum_{c,dy,dx} thr[c, 2oh+dy, 2ow+dx] )
//   folded[c,y,x] = thr[c,y,x] * sum_{(oh,ow) covering (y,x)} gate[oh,ow]
// grid = N blocks, 192 threads (one per (y,x)).
// ---------------------------------------------------------------------------
__global__ void gate_fold_kernel(const float* __restrict__ thr,
                                 float* __restrict__ fold) {
    __shared__ float thr_s[C_IN * PP * QQ];      // 576
    __shared__ float gate_s[OH * OW];            // 48

    const int n = blockIdx.x;
    const int t = threadIdx.x;                   // [0,192)

#pragma unroll
    for (int i = 0; i < 3; ++i)
        thr_s[t + i * 192] = thr[n * FLATK + t + i * 192];
    __syncthreads();

    if (t < OH * OW) {
        const int oh = t / OW, ow = t % OW;
        float s = 0.0f;
#pragma unroll
        for (int c = 0; c < C_IN; ++c) {
#pragma unroll
            for (int dy = -1; dy <= 1; ++dy) {
                const int y = 2 * oh + dy;
                if (y < 0 || y >= PP) continue;
#pragma unroll
                for (int dx = -1; dx <= 1; ++dx) {
                    const int xx = 2 * ow + dx;
                    if (xx < 0 || xx >= QQ) continue;
                    s += thr_s[(c * PP + y) * QQ + xx];
                }
            }
        }
        const float z = s * (GSCALE / 27.0f);
        gate_s[t] = 1.0f / (1.0f + __expf(-z));
    }
    __syncthreads();

    const int y = t / QQ;                        // [0,16)
    const int xx = t % QQ;                       // [0,12)

    int ohs[2]; int noh = 0;
    if ((y & 1) == 0) { ohs[noh++] = y >> 1; }
    else {
        ohs[noh++] = (y - 1) >> 1;
        if (y < 2 * OH - 1) ohs[noh++] = (y + 1) >> 1;
    }
    int ows[2]; int now = 0;
    if ((xx & 1) == 0) { ows[noh > 0 ? now++ : now++] = xx >> 1; }
    else {
        ows[now++] = (xx - 1) >> 1;
        if (xx < 2 * OW - 1) ows[now++] = (xx + 1) >> 1;
    }

    float G = 0.0f;
    for (int i = 0; i < noh; ++i)
        for (int j = 0; j < now; ++j)
            G += gate_s[ohs[i] * OW + ows[j]];

#pragma unroll
    for (int c = 0; c < C_IN; ++c) {
        const int idx = c * (PP * QQ) + t;       // (c, y, x) flattened
        fold[n * FLATK + idx] = thr_s[idx] * G;
    }
}

// ---------------------------------------------------------------------------
// Kernel 3: out(64x512) = fold(64x576) @ W(576x512) + bias, via
// V_WMMA_F32_16X16X4_F32.  One wave per 16x16 output tile; 128 waves total.
// A 16x4 lane layout: lanes 0-15 K={0,1}, lanes 16-31 K={2,3}; B symmetric.
// C/D: VGPR r -> M = r + 8*(lane>=16), N = lane & 15.
// grid = 16 blocks x 256 threads (8 waves/block).
// ---------------------------------------------------------------------------
__global__ void wmma_gemm_kernel(const float* __restrict__ fold,
                                 const float* __restrict__ W,
                                 const float* __restrict__ bias,
                                 float* __restrict__ out) {
    const int wave  = blockIdx.x * 8 + (threadIdx.x >> 5); // [0,128)
    const int lane  = threadIdx.x & 31;
    const int mt    = wave >> 5;                           // [0,4)
    const int nt    = wave & 31;                           // [0,32)
    const int mbase = mt * 16;
    const int nbase = nt * 16;
    const int half  = lane >> 4;                           // 0|1
    const int lo    = lane & 15;

    v8f acc = {};
    const float* arow = fold + (size_t)(mbase + lo) * FLATK + 2 * half;
    const float* bcol = W + (size_t)(2 * half) * NOUT + nbase + lo;

#pragma unroll 4
    for (int k0 = 0; k0 < FLATK; k0 += 4) {
        v2f a = *(const v2f*)(arow + k0);                  // K, K+1 (8B aligned)
        v2f b;
        b.x = bcol[(size_t)k0 * NOUT];
        b.y = bcol[(size_t)k0 * NOUT + NOUT];
        acc = __builtin_amdgcn_wmma_f32_16x16x4_f32(
            /*neg_a=*/false, a, /*neg_b=*/false, b,
            /*c_mod=*/(short)0, acc, /*reuse_a=*/false, /*reuse_b=*/false);
    }

    const float bv = bias[nbase + lo];
#pragma unroll
    for (int r = 0; r < 8; ++r) {
        const int m = mbase + r + 8 * half;
        out[(size_t)m * NOUT + nbase + lo] = acc[r] + bv;
    }
}

// ---------------------------------------------------------------------------
extern "C" void kernel_launch(void* const* d_in, const int* in_sizes, int n_in,
                              void* d_out, int out_size, void* d_ws, size_t ws_size,
                              hipStream_t stream) {
    (void)in_sizes; (void)n_in; (void)out_size; (void)ws_size;
    const float* x    = (const float*)d_in[0];
    const float* W    = (const float*)d_in[1];
    const float* bias = (const float*)d_in[2];
    float*       out  = (float*)d_out;

    float* thr  = (float*)d_ws;                        // 64*576 f32
    float* fold = thr + NB * FLATK;                    // 64*576 f32

    pool_thresh_kernel<<<NB * C_IN * PP, 384, 0, stream>>>(x, thr);
    gate_fold_kernel<<<NB, 192, 0, stream>>>(thr, fold);
    wmma_gemm_kernel<<<16, 256, 0, stream>>>(fold, W, bias, out);
}